// ModelNew_73315091744103
// MI455X (gfx1250) — compile-verified
//
#include <hip/hip_runtime.h>
#include <stdint.h>

typedef __attribute__((ext_vector_type(2))) float v2f;
typedef __attribute__((ext_vector_type(8))) float v8f;

#define INPUT_SIZE  8192
#define HIDDEN_SIZE 8192
#define BATCH       2048
#define SCALE_OVER_DIV 0.75f      // SCALING_FACTOR / DIVISOR = 1.5 / 2.0

// ---- Kernel 1 tiling ----
#define COLS_PER_BLK   128
#define ROWS_PER_BLK   256
#define SLAB_ROWS      16
#define NSLAB          (ROWS_PER_BLK / SLAB_ROWS)          // 16
#define LDS_STRIDE     144   // floats; pad so K and K+2 rows hit disjoint banks
#define COL_GRPS       (INPUT_SIZE / COLS_PER_BLK)         // 64
#define ROW_CHUNKS     (HIDDEN_SIZE / ROWS_PER_BLK)        // 32
#define WAVES_PER_BLK  8

// Issue one 16x128 fp32 slab (8 KB) of W into LDS using CDNA5 async
// global->LDS b128 transfers. Wave w's lane L fetches the contiguous 16-byte
// chunk at (row = w [+8], col4 = L) -> each instruction moves one contiguous
// 512 B row segment. Tracked by ASYNCcnt, no VGPR round-trip.
__device__ __forceinline__ void async_load_slab(const float* __restrict__ W,
                                                long h_base, int cb0,
                                                float* dst /* LDS */) {
  const int t = threadIdx.x;
#pragma unroll
  for (int rep = 0; rep < 2; ++rep) {
    const int idx = t + rep * 256;        // 512 float4s per slab
    const int row = idx >> 5;             // 32 float4 per 128-col row
    const int c4  = idx & 31;
    const float* g = W + (h_base + row) * (long)INPUT_SIZE + cb0 + c4 * 4;
    const unsigned lds = (unsigned)(uintptr_t)(dst + row * LDS_STRIDE + c4 * 4);
    asm volatile("global_load_async_to_lds_b128 %0, %1, off"
                 :: "v"(lds), "v"(g) : "memory");
  }
}

// ---------------------------------------------------------------------------
// Kernel 1: column sums of W via V_WMMA_F32_16X16X4_F32 with all-ones A,
// fed from an async-double-buffered LDS slab.
// D = 1(16x4) * B(4x16) + C  => every row of D = column sums of the B tile.
// ---------------------------------------------------------------------------
__global__ __launch_bounds__(256) void colsum_wmma_kernel(
    const float* __restrict__ W, float* __restrict__ partial) {
  __shared__ float buf[2][SLAB_ROWS * LDS_STRIDE];   // 2 x 9 KB

  const int colgrp = blockIdx.x % COL_GRPS;
  const int rowgrp = blockIdx.x / COL_GRPS;
  const int cb0 = colgrp * COLS_PER_BLK;
  const long h0 = (long)rowgrp * ROWS_PER_BLK;

  const int wave = threadIdx.x >> 5;
  const int lane = threadIdx.x & 31;
  const int n  = lane & 15;                 // column within 16-wide strip
  const int kb = (lane >> 4) << 1;          // K base: 0 (lanes 0-15) / 2 (16-31)
  const int cw = wave * 16;                 // this wave's strip within 128 cols

  v2f a; a.x = 1.0f; a.y = 1.0f;            // ones A-matrix (16x4)
  v8f c = {};

  async_load_slab(W, h0, cb0, &buf[0][0]);  // prologue: slab 0 in flight

  for (int s = 0; s < NSLAB; ++s) {
    if (s + 1 < NSLAB) {
      async_load_slab(W, h0 + (long)(s + 1) * SLAB_ROWS, cb0, &buf[(s + 1) & 1][0]);
      asm volatile("s_wait_asynccnt 0x2" ::: "memory");  // slab s landed
    } else {
      asm volatile("s_wait_asynccnt 0x0" ::: "memory");
    }
    __syncthreads();                        // slab s visible to all waves

    const float* sb = &buf[s & 1][0];
#pragma unroll
    for (int kk = 0; kk < SLAB_ROWS / 4; ++kk) {      // 4 WMMAs per slab
      v2f b;
      b.x = sb[(kk * 4 + kb)     * LDS_STRIDE + cw + n];
      b.y = sb[(kk * 4 + kb + 1) * LDS_STRIDE + cw + n];
      c = __builtin_amdgcn_wmma_f32_16x16x4_f32(
          /*neg_a=*/false, a, /*neg_b=*/false, b,
          /*c_mod=*/(short)0, c, /*reuse_a=*/false, /*reuse_b=*/false);
    }
    __syncthreads();                        // done reading before overwrite
  }

  // Row M=0 of D lives in c[0] on lanes 0-15 (N = lane); all rows are equal.
  if (lane < 16) {
    partial[(long)rowgrp * INPUT_SIZE + cb0 + cw + n] = c[0];
  }
}

// ---------------------------------------------------------------------------
// Kernel 2: fold 32 row-chunk partials into wsum[i] (32 KB, stays in L2).
// ---------------------------------------------------------------------------
__global__ __launch_bounds__(256) void fold_partials_kernel(
    const float* __restrict__ partial, float* __restrict__ wsum) {
  const int i = blockIdx.x * blockDim.x + threadIdx.x;
  if (i < INPUT_SIZE) {
    float s = 0.0f;
#pragma unroll
    for (int cix = 0; cix < ROW_CHUNKS; ++cix) s += partial[cix * INPUT_SIZE + i];
    wsum[i] = s;
  }
}

// ---------------------------------------------------------------------------
// Kernel 3: out[b] = 0.75 * dot(x[b,:], wsum).  One block per batch row,
// float4 global loads, wave32 shuffle reduction + LDS across the 8 waves.
// ---------------------------------------------------------------------------
__global__ __launch_bounds__(256) void rowdot_kernel(
    const float* __restrict__ x, const float* __restrict__ wsum,
    float* __restrict__ out) {
  const int b = blockIdx.x;
  const float4* xr = reinterpret_cast<const float4*>(x + (long)b * INPUT_SIZE);
  const float4* wr = reinterpret_cast<const float4*>(wsum);

  float acc = 0.0f;
  for (int idx = threadIdx.x; idx < INPUT_SIZE / 4; idx += blockDim.x) {
    float4 xv = xr[idx];
    float4 wv = wr[idx];
    acc += xv.x * wv.x + xv.y * wv.y + xv.z * wv.z + xv.w * wv.w;
  }

  for (int off = 16; off > 0; off >>= 1)
    acc += __shfl_down(acc, off, 32);

  __shared__ float smem[WAVES_PER_BLK];
  const int lane = threadIdx.x & 31;
  const int w    = threadIdx.x >> 5;
  if (lane == 0) smem[w] = acc;
  __syncthreads();
  if (threadIdx.x == 0) {
    float t = 0.0f;
#pragma unroll
    for (int k = 0; k < WAVES_PER_BLK; ++k) t += smem[k];
    out[b] = t * SCALE_OVER_DIV;
  }
}

// ---------------------------------------------------------------------------
extern "C" void kernel_launch(void* const* d_in, const int* in_sizes, int n_in,
                              void* d_out, int out_size, void* d_ws, size_t ws_size,
                              hipStream_t stream) {
  (void)in_sizes; (void)n_in; (void)out_size; (void)ws_size;
  const float* x = (const float*)d_in[0];   // [BATCH, INPUT_SIZE] fp32
  const float* W = (const float*)d_in[1];   // [HIDDEN_SIZE, INPUT_SIZE] fp32
  float* out = (float*)d_out;               // [BATCH, 1] fp32

  // Workspace: partial[32][8192] (1 MB) then wsum[8192] (32 KB).
  float* partial = (float*)d_ws;
  float* wsum    = partial + (long)ROW_CHUNKS * INPUT_SIZE;

  colsum_wmma_kernel<<<COL_GRPS * ROW_CHUNKS, 256, 0, stream>>>(W, partial);
  fold_partials_kernel<<<INPUT_SIZE / 256, 256, 0, stream>>>(partial, wsum);
  rowdot_kernel<<<BATCH, 256, 0, stream>>>(x, wsum, out);
}